// CameraParamLoss_79156247265344
// MI455X (gfx1250) — compile-verified
//
#include <hip/hip_runtime.h>
#include <hip/hip_bf16.h>
#include <stdint.h>

#define NJ   42
#define EPSL 1e-4f
#define SPB  64            // samples (== threads) per block
#define JF   (NJ * 3)      // 126 floats/sample (joints_gt)
#define LF   (NJ * 2)      // 84 floats/sample (joint_loc_2d_gt)
#define VF   (NJ)          // 42 floats/sample (joint_valid)

typedef __attribute__((ext_vector_type(4))) unsigned int tdm_g0_t;
typedef __attribute__((ext_vector_type(8))) int          tdm_g1_t;
typedef __attribute__((ext_vector_type(4))) int          tdm_g23_t;
typedef __attribute__((ext_vector_type(8))) int          tdm_g4_t;

// Issue a 1-D TDM copy of `nelem` f32 elements from global -> LDS.
// D# group0: count=1 (user), lds_addr, global_addr[56:0], type=2.
// D# group1: data_size=4B (code 2), tensor_dim0=nelem, tile_dim0=nelem,
//            tile_dim1=0 (1-D tile), tensor_dim0_stride=nelem.
__device__ __forceinline__ void tdm_copy_f32(unsigned lds_byte,
                                             const float* gsrc,
                                             unsigned nelem) {
  unsigned long long ga = (unsigned long long)(uintptr_t)gsrc;
  tdm_g0_t g0;
  g0.x = 1u;                                   // count=1, is_restore=0, gather off
  g0.y = lds_byte;                             // lds_addr (bytes)
  g0.z = (unsigned)ga;                         // global_addr[31:0]
  g0.w = (unsigned)((ga >> 32) & 0x01FFFFFFu)  // global_addr[56:32]
         | (2u << 30);                         // type = 2 ("image")
  tdm_g1_t g1 = {
    (int)(2u << 16),                           // wg_mask=0, data_size=4B, no flags
    (int)((nelem & 0xFFFFu) << 16),            // tensor_dim0[15:0]  (bits 63:48)
    (int)((nelem >> 16) & 0xFFFFu),            // tensor_dim0[31:16]; tensor_dim1 lo=0
    (int)((nelem & 0xFFFFu) << 16),            // tensor_dim1 hi=0; tile_dim0 (127:112)
    0,                                         // tile_dim1=0 (1-D), tile_dim2=0
    (int)nelem,                                // tensor_dim0_stride[31:0]
    0,                                         // stride hi / tensor_dim1_stride lo
    0
  };
  tdm_g23_t gz = {0, 0, 0, 0};                 // 2-D form: groups 2/3 unused
  tdm_g4_t  gz8 = {0, 0, 0, 0, 0, 0, 0, 0};    // trailing group (6-arg form), unused
  __builtin_amdgcn_tensor_load_to_lds(g0, g1, gz, gz, gz8, 0);
}

__global__ __launch_bounds__(SPB)
void camparam_loss_kernel(const float* __restrict__ joints_gt,   // N*42*3
                          const float* __restrict__ joint_valid, // N*42
                          const float* __restrict__ loc2d,       // N*42*2
                          const float* __restrict__ campred,     // 6*N*3
                          const float* __restrict__ rtv,         // N
                          float* __restrict__ out,               // 21*N
                          int N) {
  __shared__ float smem[SPB * (JF + LF + VF)];   // 63 KB
  float* sJ = smem;
  float* sL = smem + SPB * JF;
  float* sV = smem + SPB * (JF + LF);

  const int block0 = blockIdx.x * SPB;
  const int nsamp  = min(SPB, N - block0);
  if (nsamp <= 0) return;

  // Wave 0 issues the three TDM DMA loads; the workgroup barrier (after a
  // TENSORcnt wait in wave 0) publishes the LDS tile to the second wave.
  if (threadIdx.x < 32) {
    tdm_copy_f32((unsigned)(uintptr_t)sJ,
                 joints_gt + (size_t)block0 * JF, (unsigned)(nsamp * JF));
    tdm_copy_f32((unsigned)(uintptr_t)sL,
                 loc2d + (size_t)block0 * LF, (unsigned)(nsamp * LF));
    tdm_copy_f32((unsigned)(uintptr_t)sV,
                 joint_valid + (size_t)block0 * VF, (unsigned)(nsamp * VF));
  }

  const int  n      = block0 + (int)threadIdx.x;
  const bool active = (n < N);
  const int  nc     = active ? n : block0;     // clamped index for safe prefetch

  // Independent global loads issued *before* the DMA wait so their HBM
  // latency overlaps the TDM transfer. All are stream-once -> nontemporal.
  const float r = __builtin_nontemporal_load(rtv + nc);
  float pred[6][3];
#pragma unroll
  for (int p = 0; p < 6; ++p) {
    const float* cp = campred + ((size_t)p * N + nc) * 3;
#pragma unroll
    for (int k = 0; k < 3; ++k)
      pred[p][k] = __builtin_nontemporal_load(cp + k);
  }

  if (threadIdx.x < 32) __builtin_amdgcn_s_wait_tensorcnt(0);
  __syncthreads();
  if (!active) return;

  const float* J = sJ + threadIdx.x * JF;
  const float* L = sL + threadIdx.x * LF;
  const float* V = sV + threadIdx.x * VF;

  float a = 0.f, bx = 0.f, by = 0.f, c = 0.f, d = 0.f, e = 0.f, f = 0.f;
#pragma unroll
  for (int j = 0; j < NJ; ++j) {
    float v = V[j];
    if (j >= 21) v *= r;
    const float x = J[j * 3 + 0] * 1e-3f;
    const float y = J[j * 3 + 1] * 1e-3f;
    const float u = L[j * 2 + 0];
    const float w = L[j * 2 + 1];
    a  = fmaf(v, fmaf(x, x, y * y), a);
    bx = fmaf(v, x, bx);
    by = fmaf(v, y, by);
    c += v;
    d  = fmaf(v, fmaf(x, u, y * w), d);
    e  = fmaf(v, u, e);
    f  = fmaf(v, w, f);
  }

  // Closed-form solve of (AtA + eps I) p = Atb with
  // AtA = [[a, bx, by], [bx, c, 0], [by, 0, c]].
  const float m     = c + EPSL;
  const float A00   = a + EPSL;
  const float denom = fmaf(A00, m, -fmaf(bx, bx, by * by));
  const float num   = fmaf(d,   m, -fmaf(bx, e,  by * f));
  const float s     = num / denom;
  const float inv_m = 1.0f / m;
  const float t1    = (e - bx * s) * inv_m;
  const float t2    = (f - by * s) * inv_m;
  const float cv    = (c > 0.f) ? 1.f : 0.f;

  // Output layout (flat, return order):
  //   loss_scale (6,N,1) | loss_trans (6,N,2) | cam_param_gt (1,N,3)
  float* outScale = out;
  float* outTrans = out + 6 * (size_t)N;
  float* outGt    = out + 18 * (size_t)N;

  __builtin_nontemporal_store(s,  outGt + (size_t)n * 3 + 0);
  __builtin_nontemporal_store(t1, outGt + (size_t)n * 3 + 1);
  __builtin_nontemporal_store(t2, outGt + (size_t)n * 3 + 2);

#pragma unroll
  for (int p = 0; p < 6; ++p) {
    __builtin_nontemporal_store(fabsf(pred[p][0] - s) * cv,
                                outScale + (size_t)p * N + n);
    const size_t tb = ((size_t)p * N + n) * 2;
    __builtin_nontemporal_store(fabsf(pred[p][1] - t1) * cv, outTrans + tb + 0);
    __builtin_nontemporal_store(fabsf(pred[p][2] - t2) * cv, outTrans + tb + 1);
  }
}

extern "C" void kernel_launch(void* const* d_in, const int* in_sizes, int n_in,
                              void* d_out, int out_size, void* d_ws, size_t ws_size,
                              hipStream_t stream) {
  (void)n_in; (void)d_ws; (void)ws_size; (void)out_size;
  const float* joints_gt   = (const float*)d_in[0];  // (N,42,3)
  const float* joint_valid = (const float*)d_in[1];  // (N,42)
  const float* loc2d       = (const float*)d_in[2];  // (N,42,2)
  const float* campred     = (const float*)d_in[3];  // (6,N,3)
  const float* rtv         = (const float*)d_in[4];  // (N,1)
  // d_in[5] = rel_trans_pred: unused by the reference.
  float* out = (float*)d_out;

  const int N = in_sizes[0] / (NJ * 3);
  const int blocks = (N + SPB - 1) / SPB;
  camparam_loss_kernel<<<blocks, SPB, 0, stream>>>(
      joints_gt, joint_valid, loc2d, campred, rtv, out, N);
}